// GPM_6854767804861
// MI455X (gfx1250) — compile-verified
//
#include <hip/hip_runtime.h>
#include <hip/hip_bf16.h>
#include <math.h>
#include <stdint.h>

// ---------------- problem constants ----------------
#define BB    16          // batch
#define CIN   3           // obs channels
#define NN    3000        // nodes
#define TT    50          // time
#define EE    96000       // edges
#define RR    3           // relations
#define SS    500         // selected nodes
#define F     43          // feature dim
#define FPAD  48          // padded feature dim (multiple of 16)
#define MROWS (BB*NN)     // 48000 GEMM rows
#define WMAT  (FPAD*FPAD) // 2304 floats per packed matrix
#define WIMGN (4*WMAT + FPAD) // 9264 floats: 4 packed matrices + bias

typedef float v2f __attribute__((ext_vector_type(2)));
typedef float v8f __attribute__((ext_vector_type(8)));
typedef unsigned int u32x4 __attribute__((ext_vector_type(4)));
typedef int i32x4 __attribute__((ext_vector_type(4)));
typedef int i32x8 __attribute__((ext_vector_type(8)));

#if __has_builtin(__builtin_amdgcn_tensor_load_to_lds)
#define HAVE_TDM 1
#else
#define HAVE_TDM 0
#endif

// =====================================================================
// 1) Temporal feature extraction (unchanged from round 1; conv stacks
//    + long-max, obs tile staged through LDS, one thread per (b,n)).
// =====================================================================
__global__ __launch_bounds__(64) void temporal_kernel(
    const float* __restrict__ obs,
    const float* __restrict__ ws1, const float* __restrict__ bs1,
    const float* __restrict__ ws2, const float* __restrict__ bs2,
    const float* __restrict__ wm1, const float* __restrict__ bm1,
    const float* __restrict__ wm2, const float* __restrict__ bm2,
    float* __restrict__ X0)
{
    __shared__ float lds[CIN * TT * 64];   // 37.5 KB
    const int b  = blockIdx.y;
    const int n0 = blockIdx.x * 64;

    for (int i = threadIdx.x; i < CIN * 64 * TT; i += 64) {
        int c  = i / (64 * TT);
        int nl = (i / TT) % 64;
        int t  = i % TT;
        int n  = n0 + nl; if (n >= NN) n = NN - 1;
        lds[(c * TT + t) * 64 + nl] = obs[(((size_t)b * CIN + c) * NN + n) * TT + t];
    }
    __syncthreads();

    const int tid = threadIdx.x;
#define OBS(c, t) lds[((c) * TT + (t)) * 64 + tid]

    float s_acc[20];
#pragma unroll
    for (int o = 0; o < 20; ++o) s_acc[o] = bs2[o];
#pragma unroll 1
    for (int t = 0; t < 48; ++t) {
        float h1[3];
#pragma unroll
        for (int co = 0; co < 3; ++co) {
            float v = bs1[co];
#pragma unroll
            for (int ci = 0; ci < 3; ++ci)
#pragma unroll
                for (int kt = 0; kt < 3; ++kt)
                    v += ws1[(co * 3 + ci) * 3 + kt] * OBS(ci, t + kt);
            h1[co] = v > 0.f ? v : 0.f;
        }
#pragma unroll
        for (int o = 0; o < 20; ++o)
#pragma unroll
            for (int co = 0; co < 3; ++co)
                s_acc[o] += ws2[(o * 3 + co) * 48 + t] * h1[co];
    }
#pragma unroll
    for (int o = 0; o < 20; ++o) s_acc[o] = s_acc[o] > 0.f ? s_acc[o] : 0.f;

    float m_acc[20];
#pragma unroll
    for (int o = 0; o < 20; ++o) m_acc[o] = bm2[o];
#pragma unroll 1
    for (int t = 0; t < 30; ++t) {
        float h2[3];
#pragma unroll
        for (int co = 0; co < 3; ++co) {
            float v = bm1[co];
#pragma unroll
            for (int ci = 0; ci < 3; ++ci)
#pragma unroll 1
                for (int kt = 0; kt < 21; ++kt)
                    v += wm1[(co * 3 + ci) * 21 + kt] * OBS(ci, t + kt);
            h2[co] = v > 0.f ? v : 0.f;
        }
#pragma unroll
        for (int o = 0; o < 20; ++o)
#pragma unroll
            for (int co = 0; co < 3; ++co)
                m_acc[o] += wm2[(o * 3 + co) * 30 + t] * h2[co];
    }
#pragma unroll
    for (int o = 0; o < 20; ++o) m_acc[o] = m_acc[o] > 0.f ? m_acc[o] : 0.f;

    float lmax[3];
#pragma unroll
    for (int c = 0; c < 3; ++c) {
        float mv = -1e30f;
#pragma unroll 1
        for (int t = 0; t < TT; ++t) mv = fmaxf(mv, OBS(c, t));
        lmax[c] = mv > 0.f ? mv : 0.f;
    }
#undef OBS

    const int n = n0 + tid;
    if (n < NN) {
        float* row = X0 + ((size_t)b * NN + n) * FPAD;
#pragma unroll
        for (int o = 0; o < 20; ++o) row[o]      = s_acc[o];
#pragma unroll
        for (int o = 0; o < 20; ++o) row[20 + o] = m_acc[o];
#pragma unroll
        for (int c = 0; c < 3; ++c)  row[40 + c] = lmax[c];
#pragma unroll
        for (int c = 43; c < FPAD; ++c) row[c] = 0.f;
    }
}

// =====================================================================
// 2) Weight packer: builds the exact LDS image for the fused GEMM.
//    Pair-interleaved: WIMG[m*2304 + (p*48+n)*2 + j] = W_m[2p+j][n]
//    (so each WMMA B-fragment is ONE aligned ds_load_b64).
//    Bias (padded) appended at offset 9216.
// =====================================================================
__global__ void prep_kernel(const float* __restrict__ root, const float* __restrict__ rel,
                            const float* __restrict__ bias, float* __restrict__ WIMG)
{
    int i = blockIdx.x * blockDim.x + threadIdx.x;
    if (i < 4 * WMAT) {
        int m  = i / WMAT;
        int q  = i - m * WMAT;
        int j  = q & 1;
        int pn = q >> 1;
        int p  = pn / FPAD;
        int n  = pn - p * FPAD;
        int k  = 2 * p + j;
        float v = 0.f;
        if (k < F && n < F)
            v = (m == 0) ? root[k * F + n] : rel[(((m - 1) * F) + k) * F + n];
        WIMG[i] = v;
    } else if (i < WIMGN) {
        int n = i - 4 * WMAT;
        WIMG[i] = (n < F) ? bias[n] : 0.f;
    }
}

// =====================================================================
// 3) Fused WMMA f32 GEMM:  from one pass over X compute
//      OUT = X @ root + bias      (matrix 0)
//      YR[r] = X @ rel_r          (matrices 1..3)
//    8 waves/block; each wave owns a 16-row slab and 12 C-tiles
//    (4 matrices x 3 N-tiles), issuing 12 V_WMMA_F32_16X16X4_F32
//    per K-step. Weight image staged Global->LDS via the Tensor Data
//    Mover (TENSORcnt), cooperative-load fallback otherwise.
// =====================================================================
__global__ __launch_bounds__(256) void gemm48x4_kernel(
    const float* __restrict__ X,
    const float* __restrict__ WIMG,   // packed image, WIMGN floats
    float* __restrict__ OUT,
    float* __restrict__ YR)           // 3 x MROWS x FPAD
{
    __shared__ float Wl[WIMGN];

#if HAVE_TDM
    if (threadIdx.x < 32) {           // wave 0 issues the DMA
        unsigned long long ga = (unsigned long long)(uintptr_t)WIMG;
        unsigned int ldsa = (unsigned int)(uintptr_t)(void*)&Wl[0];
        u32x4 g0;
        g0[0] = 1u;                                              // count=1
        g0[1] = ldsa;                                            // lds_addr
        g0[2] = (unsigned int)(ga & 0xFFFFFFFFu);                // global_addr lo
        g0[3] = (unsigned int)((ga >> 32) & 0x1FFFFFFu) | (2u << 30); // addr hi | type=2
        i32x8 g1;
        g1[0] = 0x00020000;               // data_size = 4B, no multicast
        g1[1] = (int)((unsigned)WIMGN << 16);   // tensor_dim0[15:0]
        g1[2] = (int)(1u << 16);          // tensor_dim0 hi=0, tensor_dim1=1
        g1[3] = (int)((unsigned)WIMGN << 16);   // tile_dim0 = WIMGN
        g1[4] = 1;                        // tile_dim1=1, tile_dim2=0
        g1[5] = WIMGN;                    // tensor_dim0_stride lo
        g1[6] = 0; g1[7] = 0;
        i32x4 gz = {0, 0, 0, 0};
#if defined(__clang_major__) && (__clang_major__ >= 23)
        i32x8 gz8 = {0, 0, 0, 0, 0, 0, 0, 0};
        __builtin_amdgcn_tensor_load_to_lds(g0, g1, gz, gz, gz8, 0);
#else
        __builtin_amdgcn_tensor_load_to_lds(g0, g1, gz, gz, 0);
#endif
        __builtin_amdgcn_s_wait_tensorcnt(0);
    }
    __syncthreads();
#else
    for (int i = threadIdx.x; i < WIMGN; i += 256) Wl[i] = WIMG[i];
    __syncthreads();
#endif

    const int wave = threadIdx.x >> 5;        // wave32
    const int lane = threadIdx.x & 31;
    const int lo   = lane & 15;
    const int hi   = lane >> 4;
    const int mbase = (blockIdx.x * 8 + wave) * 16;

    // 12 accumulator tiles: acc[mat][ntile]
    v8f acc[4][3];
#pragma unroll
    for (int t = 0; t < 3; ++t) {
        float bv = Wl[4 * WMAT + t * 16 + lo];   // bias only for matrix 0
#pragma unroll
        for (int j = 0; j < 8; ++j) acc[0][t][j] = bv;
#pragma unroll
        for (int m = 1; m < 4; ++m)
#pragma unroll
            for (int j = 0; j < 8; ++j) acc[m][t][j] = 0.f;
    }

    const float* arow = X + ((size_t)mbase + lo) * FPAD;
    __builtin_prefetch(arow, 0, 0);           // pull A rows early (global_prefetch_b8)

#pragma unroll 1
    for (int k0 = 0; k0 < FPAD; k0 += 4) {
        const int ka = k0 + 2 * hi;           // even
        v2f a = *(const v2f*)&arow[ka];       // A frag: K=ka, K=ka+1
        const int pbase = (ka >> 1) * FPAD;   // pair row base
#pragma unroll
        for (int m = 0; m < 4; ++m) {
#pragma unroll
            for (int t = 0; t < 3; ++t) {
                v2f b = *(const v2f*)&Wl[m * WMAT + (pbase + t * 16 + lo) * 2];
                acc[m][t] = __builtin_amdgcn_wmma_f32_16x16x4_f32(
                    false, a, false, b, (short)0, acc[m][t], false, false);
            }
        }
    }

    // C/D layout: VGPR j -> (M=j, N=lane) lanes0-15 ; (M=j+8, N=lane-16)
#pragma unroll
    for (int m = 0; m < 4; ++m) {
        float* yrow = (m == 0) ? (OUT + (size_t)mbase * FPAD)
                               : (YR + (size_t)(m - 1) * MROWS * FPAD + (size_t)mbase * FPAD);
#pragma unroll
        for (int j = 0; j < 8; ++j) {
            const int r = j + 8 * hi;
            yrow[r * FPAD + lo]      = acc[m][0][j];
            yrow[r * FPAD + lo + 16] = acc[m][1][j];
            yrow[r * FPAD + lo + 32] = acc[m][2][j];
        }
    }
}

// =====================================================================
// 4) Edge aggregation helpers
// =====================================================================
__global__ void zero_kernel(float* __restrict__ p, long n)
{
    long i = (long)blockIdx.x * blockDim.x + threadIdx.x;
    if (i < n) p[i] = 0.f;
}

__global__ void cnt_kernel(const int* __restrict__ et, const int* __restrict__ dst,
                           float* __restrict__ cnt)
{
    int e = blockIdx.x * blockDim.x + threadIdx.x;
    if (e < EE) atomicAdd(&cnt[et[e] * NN + dst[e]], 1.0f);
}

__global__ void inv_kernel(const float* __restrict__ cnt, float* __restrict__ inv)
{
    int i = blockIdx.x * blockDim.x + threadIdx.x;
    if (i < RR * NN) inv[i] = 1.0f / fmaxf(cnt[i], 1.0f);
}

// Pre-scaled one-pass scatter over ALL relations:
//   OUT[b,dst,f] += YR[type][b,src,f] * inv[type][dst]
// (valid because cnt is known before the scatter; matches
//  out += sum_r agg_r / clip(cnt_r,1) up to fp reassociation)
__global__ __launch_bounds__(256) void scatter_all_kernel(
    const float* __restrict__ YR, const int* __restrict__ src,
    const int* __restrict__ dst, const int* __restrict__ et,
    const float* __restrict__ inv, float* __restrict__ OUT)
{
    const int e = blockIdx.x;
    const int r = et[e];
    const int s = src[e], d = dst[e];
    const float sc = inv[r * NN + d];
    const float* Yb = YR + (size_t)r * MROWS * FPAD;
    for (int i = threadIdx.x; i < BB * F; i += 256) {
        const int b = i / F, f = i - b * F;
        atomicAdd(&OUT[((size_t)b * NN + d) * FPAD + f],
                  Yb[((size_t)b * NN + s) * FPAD + f] * sc);
    }
}

__global__ void leaky_kernel(float* __restrict__ P)
{
    long i = (long)blockIdx.x * blockDim.x + threadIdx.x;
    if (i < (long)MROWS * FPAD) {
        float v = P[i];
        P[i] = v > 0.f ? v : 0.01f * v;
    }
}

// =====================================================================
// 5) Final head: concat [last_stocks, temporal(43), graph(43)] at the
//    selected nodes, 1x1 conv (87->1), cash logit, softmax(501).
// =====================================================================
__global__ __launch_bounds__(512) void final_kernel(
    const float* __restrict__ X0, const float* __restrict__ H2,
    const float* __restrict__ last_action, const int* __restrict__ nodes,
    const float* __restrict__ wf, const float* __restrict__ bf,
    float* __restrict__ out)
{
    __shared__ float red[512];
    const int b = blockIdx.x, tid = threadIdx.x;

    float logit = -1e30f;
    if (tid == 0) logit = 0.f;                       // cash
    if (tid >= 1 && tid <= SS) {
        const int s = tid - 1;
        const int n = nodes[s];
        float acc = bf[0] + wf[0] * last_action[b * (SS + 1) + 1 + s];
        const float* t0 = X0 + ((size_t)b * NN + n) * FPAD;
        const float* t2 = H2 + ((size_t)b * NN + n) * FPAD;
#pragma unroll 1
        for (int c = 0; c < F; ++c) acc += wf[1 + c]  * t0[c];
#pragma unroll 1
        for (int c = 0; c < F; ++c) acc += wf[44 + c] * t2[c];
        logit = acc;
    }

    red[tid] = logit; __syncthreads();
#pragma unroll
    for (int s2 = 256; s2 > 0; s2 >>= 1) {
        if (tid < s2) red[tid] = fmaxf(red[tid], red[tid + s2]);
        __syncthreads();
    }
    const float mx = red[0]; __syncthreads();

    float ex = (tid <= SS) ? expf(logit - mx) : 0.f;
    red[tid] = ex; __syncthreads();
#pragma unroll
    for (int s2 = 256; s2 > 0; s2 >>= 1) {
        if (tid < s2) red[tid] += red[tid + s2];
        __syncthreads();
    }
    const float denom = red[0];
    if (tid <= SS) out[b * (SS + 1) + tid] = ex / denom;
}

// =====================================================================
// Host-side pipeline (13 launches total)
// =====================================================================
extern "C" void kernel_launch(void* const* d_in, const int* in_sizes, int n_in,
                              void* d_out, int out_size, void* d_ws, size_t ws_size,
                              hipStream_t stream)
{
    (void)in_sizes; (void)n_in; (void)out_size; (void)ws_size;

    const float* obs         = (const float*)d_in[0];
    const float* last_action = (const float*)d_in[1];
    const int*   edge_index  = (const int*)  d_in[2];
    const int*   edge_type   = (const int*)  d_in[3];
    const int*   nodes       = (const int*)  d_in[4];
    const float* ws1 = (const float*)d_in[5];  const float* bs1 = (const float*)d_in[6];
    const float* ws2 = (const float*)d_in[7];  const float* bs2 = (const float*)d_in[8];
    const float* wm1 = (const float*)d_in[9];  const float* bm1 = (const float*)d_in[10];
    const float* wm2 = (const float*)d_in[11]; const float* bm2 = (const float*)d_in[12];
    const float* root1 = (const float*)d_in[13]; const float* rel1 = (const float*)d_in[14];
    const float* bias1 = (const float*)d_in[15];
    const float* root2 = (const float*)d_in[16]; const float* rel2 = (const float*)d_in[17];
    const float* bias2 = (const float*)d_in[18];
    const float* wf = (const float*)d_in[19];  const float* bf = (const float*)d_in[20];

    const int* src = edge_index;
    const int* dst = edge_index + EE;

    const long MF = (long)MROWS * FPAD;          // 2,304,000 floats
    float* ws   = (float*)d_ws;
    float* X0   = ws;                            // temporal features (kept for head)
    float* X1   = X0 + MF;                       // layer-1 output
    float* X2   = X1 + MF;                       // layer-2 output
    float* YR   = X2 + MF;                       // 3 relation GEMM results
    float* WIMG = YR + 3 * MF;                   // packed weight image (LDS mirror)
    float* CNT  = WIMG + WIMGN;
    float* CINV = CNT + RR * NN;

    // ---- temporal features ----
    dim3 tgrid((NN + 63) / 64, BB);
    temporal_kernel<<<tgrid, 64, 0, stream>>>(obs, ws1, bs1, ws2, bs2,
                                              wm1, bm1, wm2, bm2, X0);

    // ---- edge counts (shared by both layers) ----
    zero_kernel<<<(RR * NN + 255) / 256, 256, 0, stream>>>(CNT, RR * NN);
    cnt_kernel<<<(EE + 255) / 256, 256, 0, stream>>>(edge_type, dst, CNT);
    inv_kernel<<<(RR * NN + 255) / 256, 256, 0, stream>>>(CNT, CINV);

    // ---- two RGCN layers ----
    const float* roots[2]  = {root1, root2};
    const float* rels[2]   = {rel1,  rel2};
    const float* biases[2] = {bias1, bias2};
    const float* Xin = X0;
    float*       Xout = X1;
    const int elem_blocks = (int)((MF + 255) / 256);
    const int gemm_blocks = MROWS / 128;         // 375, exact tiling

    for (int l = 0; l < 2; ++l) {
        prep_kernel<<<(WIMGN + 255) / 256, 256, 0, stream>>>(
            roots[l], rels[l], biases[l], WIMG);
        gemm48x4_kernel<<<gemm_blocks, 256, 0, stream>>>(Xin, WIMG, Xout, YR);
        scatter_all_kernel<<<EE, 256, 0, stream>>>(YR, src, dst, edge_type, CINV, Xout);
        leaky_kernel<<<elem_blocks, 256, 0, stream>>>(Xout);
        Xin = Xout;
        Xout = X2;
    }

    // ---- head + softmax ----
    final_kernel<<<BB, 512, 0, stream>>>(X0, X2, last_action, nodes, wf, bf,
                                         (float*)d_out);
}